// DMaSIFConvBlock_61847529062863
// MI455X (gfx1250) — compile-verified
//
#include <hip/hip_runtime.h>
#include <hip/hip_bf16.h>

typedef __attribute__((ext_vector_type(2))) float v2f;
typedef __attribute__((ext_vector_type(8))) float v8f;

#define WAVES_PER_BLOCK 8
#define BLOCK_THREADS   256

// Fused 16->16->relu->16 MLP on 16-row tiles using V_WMMA_F32_16X16X4_F32.
// A-operand (16x4 f32, 2 VGPRs): lane L holds rows M=L%16, K-pair (2*(L/16), +1).
// B-operand (4x16 f32, 2 VGPRs): lane L holds cols N=L%16, K rows (j + 2*(L/16)).
// C/D (16x16 f32, 8 VGPRs): vgpr i, lane L -> (M = i + 8*(L/16), N = L%16).
__global__ __launch_bounds__(BLOCK_THREADS) void dmasif_mlp_wmma(
    const float* __restrict__ X,
    const float* __restrict__ Wt, const float* __restrict__ bt,
    const float* __restrict__ Wa, const float* __restrict__ ba,
    const float* __restrict__ Wb, const float* __restrict__ bb,
    float* __restrict__ Y, int num_tiles)
{
    __shared__ float lds[WAVES_PER_BLOCK][16 * 16];

    const int lane  = threadIdx.x & 31;
    const int wave  = threadIdx.x >> 5;
    const int gwave = blockIdx.x * WAVES_PER_BLOCK + wave;
    const int nwav  = gridDim.x * WAVES_PER_BLOCK;

    const int o    = lane & 15;   // N (column) index for B/C/D; M (row) index for A
    const int half = lane >> 4;   // lane half -> +2 K offset (A/B), +8 M offset (C/D)

    // ---- preload weight B-operands (B[k][n] = W[n][k]) and biases ----
    v2f Bt[4], Ba[4], Bb[4];
#pragma unroll
    for (int kc = 0; kc < 4; ++kc) {
        const int k = 4 * kc + 2 * half;
        Bt[kc] = *reinterpret_cast<const v2f*>(Wt + o * 16 + k);
        Ba[kc] = *reinterpret_cast<const v2f*>(Wa + o * 16 + k);
        Bb[kc] = *reinterpret_cast<const v2f*>(Wb + o * 16 + k);
    }
    const float btv = bt[o];
    const float bav = ba[o];
    const float bbv = bb[o];

    float* tile = lds[wave];

    for (int t = gwave; t < num_tiles; t += nwav) {
        const float* Xt = X + (size_t)t * 256;

        // speculative prefetch of this wave's next tile (1 KB)
        __builtin_prefetch(X + ((size_t)t + nwav) * 256, 0, 0);

        // ---------- layer 1: H1 = X @ Wt^T + bt ----------
        v8f c;
#pragma unroll
        for (int i = 0; i < 8; ++i) c[i] = btv;
#pragma unroll
        for (int kc = 0; kc < 4; ++kc) {
            v2f a = *reinterpret_cast<const v2f*>(Xt + o * 16 + 4 * kc + 2 * half);
            c = __builtin_amdgcn_wmma_f32_16x16x4_f32(
                    false, a, false, Bt[kc], (short)0, c, false, false);
        }

        // D-layout -> row-major LDS (transpose bounce); DS in-order per wave
#pragma unroll
        for (int i = 0; i < 8; ++i)
            tile[(i + 8 * half) * 16 + o] = c[i];
        asm volatile("s_wait_dscnt 0" ::: "memory");

        // ---------- layer 2: H2 = relu(H1 @ Wa^T + ba) ----------
#pragma unroll
        for (int i = 0; i < 8; ++i) c[i] = bav;
#pragma unroll
        for (int kc = 0; kc < 4; ++kc) {
            v2f a = *reinterpret_cast<const v2f*>(tile + o * 16 + 4 * kc + 2 * half);
            c = __builtin_amdgcn_wmma_f32_16x16x4_f32(
                    false, a, false, Ba[kc], (short)0, c, false, false);
        }
#pragma unroll
        for (int i = 0; i < 8; ++i) c[i] = fmaxf(c[i], 0.0f);

        asm volatile("" ::: "memory");
#pragma unroll
        for (int i = 0; i < 8; ++i)
            tile[(i + 8 * half) * 16 + o] = c[i];
        asm volatile("s_wait_dscnt 0" ::: "memory");

        // ---------- layer 3: Y = H2 @ Wb^T + bb ----------
#pragma unroll
        for (int i = 0; i < 8; ++i) c[i] = bbv;
#pragma unroll
        for (int kc = 0; kc < 4; ++kc) {
            v2f a = *reinterpret_cast<const v2f*>(tile + o * 16 + 4 * kc + 2 * half);
            c = __builtin_amdgcn_wmma_f32_16x16x4_f32(
                    false, a, false, Bb[kc], (short)0, c, false, false);
        }
        asm volatile("" ::: "memory");

        // D-layout store: lanes 0-15 / 16-31 each write 64B-contiguous row halves
        float* Yt = Y + (size_t)t * 256;
#pragma unroll
        for (int i = 0; i < 8; ++i)
            Yt[(i + 8 * half) * 16 + o] = c[i];
    }
}

// Scalar tail for rows not covered by full 16-row tiles (unused when N % 16 == 0).
__global__ void dmasif_mlp_tail(
    const float* __restrict__ X,
    const float* __restrict__ Wt, const float* __restrict__ bt,
    const float* __restrict__ Wa, const float* __restrict__ ba,
    const float* __restrict__ Wb, const float* __restrict__ bb,
    float* __restrict__ Y, int row_start, int n_rows)
{
    int r = row_start + blockIdx.x * blockDim.x + threadIdx.x;
    if (r >= n_rows) return;
    float x[16], h1[16], h2[16];
#pragma unroll
    for (int k = 0; k < 16; ++k) x[k] = X[r * 16 + k];
#pragma unroll
    for (int o = 0; o < 16; ++o) {
        float s = bt[o];
        for (int k = 0; k < 16; ++k) s += x[k] * Wt[o * 16 + k];
        h1[o] = s;
    }
#pragma unroll
    for (int o = 0; o < 16; ++o) {
        float s = ba[o];
        for (int k = 0; k < 16; ++k) s += h1[k] * Wa[o * 16 + k];
        h2[o] = fmaxf(s, 0.0f);
    }
#pragma unroll
    for (int o = 0; o < 16; ++o) {
        float s = bb[o];
        for (int k = 0; k < 16; ++k) s += h2[k] * Wb[o * 16 + k];
        Y[r * 16 + o] = s;
    }
}

extern "C" void kernel_launch(void* const* d_in, const int* in_sizes, int n_in,
                              void* d_out, int out_size, void* d_ws, size_t ws_size,
                              hipStream_t stream) {
    // setup_inputs order: features, points, nuv, Wt, bt, Wa, ba, Wb, bb, ranges
    const float* X  = (const float*)d_in[0];
    const float* Wt = (const float*)d_in[3];
    const float* bt = (const float*)d_in[4];
    const float* Wa = (const float*)d_in[5];
    const float* ba = (const float*)d_in[6];
    const float* Wb = (const float*)d_in[7];
    const float* bb = (const float*)d_in[8];
    float* Y = (float*)d_out;

    const int n_rows = in_sizes[0] / 16;   // N = 2,000,000
    const int tiles  = n_rows / 16;        // 125,000 full 16-row tiles
    const int rem    = n_rows - tiles * 16;

    if (tiles > 0) {
        int grid = 2048;  // 16384 waves, ~7.6 tiles each; plenty to saturate HBM
        int max_blocks = (tiles + WAVES_PER_BLOCK - 1) / WAVES_PER_BLOCK;
        if (grid > max_blocks) grid = max_blocks;
        dmasif_mlp_wmma<<<grid, BLOCK_THREADS, 0, stream>>>(
            X, Wt, bt, Wa, ba, Wb, bb, Y, tiles);
    }
    if (rem > 0) {
        dmasif_mlp_tail<<<(rem + 255) / 256, 256, 0, stream>>>(
            X, Wt, bt, Wa, ba, Wb, bb, Y, tiles * 16, n_rows);
    }
}